// GCNencoder_45509473468998
// MI455X (gfx1250) — compile-verified
//
#include <hip/hip_runtime.h>

typedef __attribute__((ext_vector_type(2))) float v2f;
typedef __attribute__((ext_vector_type(8))) float v8f;

#define N_NODES 50000
#define N_EDGES 800000

// ---------------------------------------------------------------- utilities
__global__ void fill_kernel(float* __restrict__ p, float v, int n) {
    int i = blockIdx.x * blockDim.x + threadIdx.x;
    if (i < n) p[i] = v;
}

// deg[c] += ew[e]  (deg buffer pre-filled with 1.0 for the self loop)
__global__ void deg_accum_kernel(const int* __restrict__ col,
                                 const float* __restrict__ ew,
                                 float* __restrict__ deg, int E) {
    int e = blockIdx.x * blockDim.x + threadIdx.x;
    if (e < E) atomicAdd(&deg[col[e]], ew[e]);
}

// in-place: deg -> dinv = deg > 0 ? rsqrt(deg) : 0
__global__ void dinv_kernel(float* __restrict__ deg, int n) {
    int i = blockIdx.x * blockDim.x + threadIdx.x;
    if (i < n) {
        float d = deg[i];
        deg[i] = (d > 0.0f) ? rsqrtf(d) : 0.0f;
    }
}

// norm[e] = dinv[row[e]] * ew[e] * dinv[col[e]]
__global__ void norm_kernel(const int* __restrict__ row, const int* __restrict__ col,
                            const float* __restrict__ ew, const float* __restrict__ dinv,
                            float* __restrict__ norm, int E) {
    int e = blockIdx.x * blockDim.x + threadIdx.x;
    if (e < E) norm[e] = dinv[row[e]] * ew[e] * dinv[col[e]];
}

// ---------------------------------------------------- WMMA f32 dense GEMM
// H[N, DOUT] = X[N, DIN] @ W[DIN, DOUT], N multiple of 16.
// One block per 16-row tile; DOUT/16 waves, each owns a 16-col tile.
// Uses V_WMMA_F32_16X16X4_F32 with A staged through LDS.
template <int DIN, int DOUT>
__launch_bounds__(DOUT * 2)
__global__ void gemm_wmma_kernel(const float* __restrict__ X,
                                 const float* __restrict__ W,
                                 float* __restrict__ H) {
    __shared__ float As[16 * DIN];
    const int row0 = blockIdx.x * 16;
    const int lane = threadIdx.x & 31;
    const int wave = threadIdx.x >> 5;           // 0 .. DOUT/16-1
    const int col0 = wave * 16;

    // Cooperative load of the 16 x DIN A tile into LDS (coalesced along DIN).
    for (int i = threadIdx.x; i < 16 * DIN; i += DOUT * 2) {
        int r = i / DIN;
        int c = i - r * DIN;
        As[i] = X[(size_t)(row0 + r) * DIN + c];
    }
    __syncthreads();

    // ISA A layout (32-bit A 16x4): lanes 0-15 -> K=0,1 ; lanes 16-31 -> K=2,3.
    const int m  = lane & 15;
    const int kb = (lane >> 4) * 2;

    v8f acc = {};
    for (int k = 0; k < DIN; k += 4) {
        v2f a, b;
        a.x = As[m * DIN + k + kb];
        a.y = As[m * DIN + k + kb + 1];
        // B (4x16) mirrored: VGPR0 = row K=kb, VGPR1 = row K=kb+1, N = lane&15.
        b.x = W[(size_t)(k + kb) * DOUT + col0 + m];
        b.y = W[(size_t)(k + kb + 1) * DOUT + col0 + m];
        acc = __builtin_amdgcn_wmma_f32_16x16x4_f32(
            /*neg_a=*/false, a, /*neg_b=*/false, b,
            /*c_mod=*/(short)0, acc, /*reuse_a=*/false, /*reuse_b=*/false);
    }

    // C/D layout: VGPR r, lane l -> M = r + 8*(l>>4), N = l&15.
    const int n     = lane & 15;
    const int mbase = (lane >> 4) * 8;
#pragma unroll
    for (int r = 0; r < 8; ++r) {
        H[(size_t)(row0 + mbase + r) * DOUT + col0 + n] = acc[r];
    }
}

// -------------------------------------------------- edge scatter (message passing)
// agg[col[e], f] += H[row[e], f] * norm[e]; one feature element per thread,
// consecutive lanes cover consecutive features of the same edge (coalesced,
// broadcast row/col/norm loads within a wave for d >= 32).
__global__ void scatter_kernel(const float* __restrict__ H,
                               const int* __restrict__ row,
                               const int* __restrict__ col,
                               const float* __restrict__ norm,
                               float* __restrict__ agg, int E, int dlog2) {
    int t = blockIdx.x * blockDim.x + threadIdx.x;
    if (t >= (E << dlog2)) return;
    int e = t >> dlog2;
    int f = t & ((1 << dlog2) - 1);
    int d = 1 << dlog2;
    float v = H[(size_t)row[e] * d + f] * norm[e];
    atomicAdd(&agg[(size_t)col[e] * d + f], v);
}

// out = relu(agg + H * dinv^2 (self loop) + bias)
__global__ void finish_kernel(const float* __restrict__ agg,
                              const float* __restrict__ H,
                              const float* __restrict__ dinv,
                              const float* __restrict__ bias,
                              float* __restrict__ out, int total, int dlog2) {
    int t = blockIdx.x * blockDim.x + threadIdx.x;
    if (t >= total) return;
    int node = t >> dlog2;
    int f = t & ((1 << dlog2) - 1);
    float di = dinv[node];
    float v = agg[t] + H[t] * di * di + bias[f];
    out[t] = (v > 0.0f) ? v : 0.0f;
}

// ---------------------------------------------------------------- launcher
extern "C" void kernel_launch(void* const* d_in, const int* in_sizes, int n_in,
                              void* d_out, int out_size, void* d_ws, size_t ws_size,
                              hipStream_t stream) {
    (void)in_sizes; (void)n_in; (void)out_size; (void)ws_size;

    const float* x  = (const float*)d_in[0];          // [50000, 256]
    const int*   ei = (const int*)d_in[1];            // [2, 800000]
    const float* ew = (const float*)d_in[2];          // [800000]
    const float* W1 = (const float*)d_in[3];
    const float* b1 = (const float*)d_in[4];
    const float* W2 = (const float*)d_in[5];
    const float* b2 = (const float*)d_in[6];
    const float* W3 = (const float*)d_in[7];
    const float* b3 = (const float*)d_in[8];

    const int* row = ei;             // sources
    const int* col = ei + N_EDGES;   // targets

    float* ws   = (float*)d_ws;
    float* dinv = ws;  ws += N_NODES;                  // deg -> dinv (in place)
    float* norm = ws;  ws += N_EDGES;
    float* bufH = ws;  ws += (size_t)N_NODES * 128;
    float* bufA = ws;  ws += (size_t)N_NODES * 128;
    float* bufX = ws;  ws += (size_t)N_NODES * 128;

    const int TPB = 256;
    const int gN  = (N_NODES + TPB - 1) / TPB;
    const int gE  = (N_EDGES + TPB - 1) / TPB;

    // degree / norm precompute
    fill_kernel<<<gN, TPB, 0, stream>>>(dinv, 1.0f, N_NODES);   // self-loop weight
    deg_accum_kernel<<<gE, TPB, 0, stream>>>(col, ew, dinv, N_EDGES);
    dinv_kernel<<<gN, TPB, 0, stream>>>(dinv, N_NODES);
    norm_kernel<<<gE, TPB, 0, stream>>>(row, col, ew, dinv, norm, N_EDGES);

    const int rowTiles = N_NODES / 16;  // 3125, exact

    // ---- layer 1: 256 -> 128
    {
        const int d = 128, dl = 7;
        gemm_wmma_kernel<256, 128><<<rowTiles, 256, 0, stream>>>(x, W1, bufH);
        int tot = N_NODES * d;
        fill_kernel<<<(tot + TPB - 1) / TPB, TPB, 0, stream>>>(bufA, 0.0f, tot);
        int etot = N_EDGES << dl;
        scatter_kernel<<<(etot + TPB - 1) / TPB, TPB, 0, stream>>>(bufH, row, col, norm, bufA, N_EDGES, dl);
        finish_kernel<<<(tot + TPB - 1) / TPB, TPB, 0, stream>>>(bufA, bufH, dinv, b1, bufX, tot, dl);
    }
    // ---- layer 2: 128 -> 64
    {
        const int d = 64, dl = 6;
        gemm_wmma_kernel<128, 64><<<rowTiles, 128, 0, stream>>>(bufX, W2, bufH);
        int tot = N_NODES * d;
        fill_kernel<<<(tot + TPB - 1) / TPB, TPB, 0, stream>>>(bufA, 0.0f, tot);
        int etot = N_EDGES << dl;
        scatter_kernel<<<(etot + TPB - 1) / TPB, TPB, 0, stream>>>(bufH, row, col, norm, bufA, N_EDGES, dl);
        finish_kernel<<<(tot + TPB - 1) / TPB, TPB, 0, stream>>>(bufA, bufH, dinv, b2, bufX, tot, dl);
    }
    // ---- layer 3: 64 -> 32 (writes d_out)
    {
        const int d = 32, dl = 5;
        gemm_wmma_kernel<64, 32><<<rowTiles, 64, 0, stream>>>(bufX, W3, bufH);
        int tot = N_NODES * d;
        fill_kernel<<<(tot + TPB - 1) / TPB, TPB, 0, stream>>>(bufA, 0.0f, tot);
        int etot = N_EDGES << dl;
        scatter_kernel<<<(etot + TPB - 1) / TPB, TPB, 0, stream>>>(bufH, row, col, norm, bufA, N_EDGES, dl);
        finish_kernel<<<(tot + TPB - 1) / TPB, TPB, 0, stream>>>(bufA, bufH, dinv, b3, (float*)d_out, tot, dl);
    }
}